// Loss_Longtail_Division_46505905881692
// MI455X (gfx1250) — compile-verified
//
#include <hip/hip_runtime.h>

#define V_DIM 10000
#define T_DIM 32
#define NT    256
#define NWAVE (NT / 32)
#define WCHUNK 1280   // floats per wave slice (last wave gets the 1040-float tail)

typedef __attribute__((address_space(3))) float lds_float;

__device__ __forceinline__ float waveMax(float v) {
    #pragma unroll
    for (int off = 16; off > 0; off >>= 1)
        v = fmaxf(v, __shfl_xor(v, off, 32));
    return v;
}
__device__ __forceinline__ float waveSum(float v) {
    #pragma unroll
    for (int off = 16; off > 0; off >>= 1)
        v += __shfl_xor(v, off, 32);
    return v;
}

__global__ __launch_bounds__(NT)
void pl_longtail_loss_kernel(const float* __restrict__ out,
                             const float* __restrict__ wbuf,  // inv weights (or raw)
                             int use_inv,
                             const int* __restrict__ target,
                             const int* __restrict__ longtail,
                             float* __restrict__ neg) {
    __shared__ __align__(16) float s_row[V_DIM];   // 40000 B
    __shared__ float s_red[NWAVE];
    __shared__ float s_m, s_S;

    const int b    = blockIdx.x;
    const int tid  = threadIdx.x;
    const int lane = tid & 31;
    const int wid  = tid >> 5;

    const float* row = out + (size_t)b * V_DIM;

    // Per-wave contiguous slice: wave w owns [start, end) of the row.
    const int start = wid * WCHUNK;
    const int end   = (start + WCHUNK < V_DIM) ? (start + WCHUNK) : V_DIM;

    // ---- Phase 0: async global -> LDS copy of this wave's slice (CDNA5 async path)
    // b128 chunks, ASYNCcnt-tracked; each wave waits only for its own data, no
    // block barrier needed before touching its own slice.
    {
        const unsigned lds_base = (unsigned)(unsigned long long)(lds_float*)s_row;
        const int c1 = end >> 2;
        for (int c = (start >> 2) + lane; c < c1; c += 32) {
            unsigned     loff = lds_base + (unsigned)(c * 16);
            const float* g    = row + c * 4;
            asm volatile("global_load_async_to_lds_b128 %0, %1, off"
                         :: "v"(loff), "v"(g) : "memory");
        }
        asm volatile("s_wait_asynccnt 0" ::: "memory");
    }

    // ---- Phase 1: scale own slice in place, local max ----
    float lmax = -3.402823466e38f;
    for (int i = start + lane; i < end; i += 32) {
        float r = s_row[i];
        float w = wbuf[i];                 // 40 KB table, L2-resident for all blocks
        float s = use_inv ? (r * w) : (r / w);
        s_row[i] = s;
        lmax = fmaxf(lmax, s);
    }
    lmax = waveMax(lmax);
    if (lane == 0) s_red[wid] = lmax;
    __syncthreads();                       // also publishes every wave's scaled slice
    if (wid == 0) {
        float x = (lane < NWAVE) ? s_red[lane] : -3.402823466e38f;
        x = waveMax(x);
        if (lane == 0) s_m = x;
    }
    __syncthreads();
    const float m = s_m;

    // ---- Phase 2: sum exp(s - m) over own slice ----
    float lsum = 0.0f;
    for (int i = start + lane; i < end; i += 32)
        lsum += __expf(s_row[i] - m);      // hw v_exp_f32 keeps the bulk pass cheap
    lsum = waveSum(lsum);
    if (lane == 0) s_red[wid] = lsum;      // safe: last s_red read fenced by barrier above
    __syncthreads();
    if (wid == 0) {
        float x = (lane < NWAVE) ? s_red[lane] : 0.0f;
        x = waveSum(x);
        if (lane == 0) s_S = x;
    }
    __syncthreads();
    const float S   = s_S;
    const float lse = m + logf(S);

    // ---- Phase 3: tail Plackett-Luce (wave 0 only; T = 32 = one wave32) ----
    if (wid == 0) {
        const int   tgt  = target[b];
        const float tsc  = s_row[tgt];             // gather from LDS
        const float expt = expf(tsc - m);

        const int   li  = longtail[(size_t)b * T_DIM + lane];
        const float msk = (li > 0) ? 1.0f : 0.0f;  // value 0 == padding
        const float ts  = s_row[li];               // gather from LDS
        const float et  = expf(ts - m) * msk;

        // inclusive suffix scan: suf[t] = sum_{j>=t} et[j]
        float suf = et;
        #pragma unroll
        for (int off = 1; off < 32; off <<= 1) {
            float n = __shfl_down(suf, off, 32);
            if (lane + off < 32) suf += n;
        }
        const float sum_tail = __shfl(suf, 0, 32); // full tail sum
        const float other    = S - expt - sum_tail; // all in exp(-m)-scaled space

        // tail_s - log(suffix + other) = ts - m - log(suf_hat + other_hat)
        const float term = (ts - m - logf(suf + other)) * msk;
        const float lpt  = waveSum(term);

        if (lane == 0)
            neg[b] = -((tsc - lse) + lpt);
    }
}

__global__ void inv_weight_kernel(const float* __restrict__ w,
                                  float* __restrict__ inv, int n) {
    int i = blockIdx.x * blockDim.x + threadIdx.x;
    if (i < n) inv[i] = 1.0f / w[i];   // exact IEEE division, done once per column
}

extern "C" void kernel_launch(void* const* d_in, const int* in_sizes, int n_in,
                              void* d_out, int out_size, void* d_ws, size_t ws_size,
                              hipStream_t stream) {
    const float* output   = (const float*)d_in[0];   // [B, V] f32
    const float* lw       = (const float*)d_in[1];   // [1, V] f32
    const int*   target   = (const int*)  d_in[2];   // [B]    i32
    const int*   longtail = (const int*)  d_in[3];   // [B, T] i32
    float*       neg      = (float*)d_out;           // [B]    f32

    const int B = in_sizes[2];

    // Precompute 1/w once (exact division), then the hot kernel multiplies.
    const float* wbuf = lw;
    int use_inv = 0;
    if (ws_size >= (size_t)V_DIM * sizeof(float)) {
        float* inv = (float*)d_ws;
        inv_weight_kernel<<<(V_DIM + NT - 1) / NT, NT, 0, stream>>>(lw, inv, V_DIM);
        wbuf = inv;
        use_inv = 1;
    }

    pl_longtail_loss_kernel<<<B, NT, 0, stream>>>(output, wbuf, use_inv,
                                                  target, longtail, neg);
}